// HybridLoss_47175920779947
// MI455X (gfx1250) — compile-verified
//
#include <hip/hip_runtime.h>
#include <hip/hip_bf16.h>

// ---------------------------------------------------------------------------
// HybridLoss for MI455X (gfx1250): fused streaming reduction.
//   seg_out (2,4,128,128,128) f32 | calib (20,4) f32 | logits same as seg |
//   y (2,1,128,128,128) i32 | weight scalar i32  ->  scalar f32 loss.
// Memory-bound: 144MB @ 23.3TB/s ~ 6.2us. One pass over all inputs.
// Cross-lane reductions done entirely with V_WMMA_F32_16X16X4_F32 (exact f32).
// ---------------------------------------------------------------------------

constexpr int V = 128 * 128 * 128;   // voxels per (batch, channel) plane
constexpr int NBATCH = 2;

typedef __attribute__((ext_vector_type(2))) float v2f;
typedef __attribute__((ext_vector_type(8))) float v8f;

// Exact f32 wave32 all-reduce using only V_WMMA_F32_16X16X4_F32 (no LDS, no
// shuffles, no dscnt waits).
//
// Step 1: A = (x, 0) per lane -> A[m][0]=x[m], A[m][2]=x[m+16]; B = ones.
//         D1[m][n] = x[m] + x[m+16]  (rowsum[m], replicated over n).
//         Layout: D1 VGPR r = rowsum[r] in lanes 0-15, rowsum[8+r] in 16-31.
// Step 2: four accumulating WMMAs, A_i = (D1[2i], D1[2i+1]), B = ones:
//         each contributes rowsum[2i]+rowsum[2i+1]+rowsum[8+2i]+rowsum[9+2i].
//         After i=0..3 every element of D = sum of all 32 lanes.
__device__ __forceinline__ float wave_allreduce(float x) {
  v2f ones; ones[0] = 1.0f; ones[1] = 1.0f;
  v2f a;    a[0] = x;       a[1] = 0.0f;
  v8f z = {};
  v8f d1 = __builtin_amdgcn_wmma_f32_16x16x4_f32(
      false, a, false, ones, (short)0, z, false, false);
  v8f acc = {};
#pragma unroll
  for (int i = 0; i < 4; ++i) {
    v2f ai; ai[0] = d1[2 * i]; ai[1] = d1[2 * i + 1];
    acc = __builtin_amdgcn_wmma_f32_16x16x4_f32(
        false, ai, false, ones, (short)0, acc, false, false);
  }
  return acc[0];  // total in every lane
}

__device__ __forceinline__ float sigmoidf_(float x) {
  float t = expf(-fabsf(x));
  return (x >= 0.0f) ? 1.0f / (1.0f + t) : t / (1.0f + t);
}

__device__ __forceinline__ float comp4(const float4& f, int j) {
  return j == 0 ? f.x : j == 1 ? f.y : j == 2 ? f.z : f.w;
}
__device__ __forceinline__ int comp4i(const int4& f, int j) {
  return j == 0 ? f.x : j == 1 ? f.y : j == 2 ? f.z : f.w;
}

__global__ void zero_hist(unsigned int* __restrict__ h) {
  if (threadIdx.x < 80) h[threadIdx.x] = 0u;
}

// Stat layout per batch (20 floats):
//  [0..3]  sum pred_c            [4..7]  sum pred_c * 1[y==c]
//  [8..11] count(y==c)           [12..15] sum sigmoid(logit_c)
//  [16..19] sum pred_c * sigmoid(logit_c)
__global__ void __launch_bounds__(256)
stats_kernel(const float* __restrict__ seg, const float* __restrict__ logit,
             const int* __restrict__ y, float* __restrict__ blockPartials,
             unsigned int* __restrict__ histG, int bpb, int iters) {
  __shared__ unsigned int hist[8 * 80];   // per-wave histogram replicas
  __shared__ float wp[8 * 20];            // per-wave reduced stats

  const int tid  = threadIdx.x;
  const int wid  = tid >> 5;
  const int lane = tid & 31;

  for (int i = tid; i < 8 * 80; i += 256) hist[i] = 0u;
  __syncthreads();

  const int b   = blockIdx.x / bpb;
  const int blk = blockIdx.x % bpb;
  const float* segB = seg   + (size_t)b * 4 * V;
  const float* logB = logit + (size_t)b * 4 * V;
  const int*   yB   = y     + (size_t)b * V;

  float sp[4]  = {0.f, 0.f, 0.f, 0.f};
  float ipr[4] = {0.f, 0.f, 0.f, 0.f};
  float ct[4]  = {0.f, 0.f, 0.f, 0.f};
  float ss[4]  = {0.f, 0.f, 0.f, 0.f};
  float sps[4] = {0.f, 0.f, 0.f, 0.f};

  unsigned int* h = hist + wid * 80;
  const int base = blk * iters * 256;     // float4-group base for this block

  for (int it = 0; it < iters; ++it) {
    const int g = base + it * 256 + tid;
    const int v = g * 4;
    float4 P[4], L[4];
#pragma unroll
    for (int cc = 0; cc < 4; ++cc) {
      P[cc] = *reinterpret_cast<const float4*>(segB + (size_t)cc * V + v);
      L[cc] = *reinterpret_cast<const float4*>(logB + (size_t)cc * V + v);
    }
    const int4 Y = *reinterpret_cast<const int4*>(yB + v);

#pragma unroll
    for (int j = 0; j < 4; ++j) {
      const int yi = comp4i(Y, j);
      float pr[4];
#pragma unroll
      for (int cc = 0; cc < 4; ++cc) {
        pr[cc] = comp4(P[cc], j);
        const float sg = sigmoidf_(comp4(L[cc], j));
        sp[cc]  += pr[cc];
        ss[cc]  += sg;
        sps[cc] += pr[cc] * sg;
        const float m = (yi == cc) ? 1.0f : 0.0f;  // one-hot, branchless
        ipr[cc] += m * pr[cc];
        ct[cc]  += m;
      }
      // softmax over channels (classes 1,2 feed the calibration bins)
      const float mx = fmaxf(fmaxf(pr[0], pr[1]), fmaxf(pr[2], pr[3]));
      const float e0 = expf(pr[0] - mx), e1 = expf(pr[1] - mx);
      const float e2 = expf(pr[2] - mx), e3 = expf(pr[3] - mx);
      const float sm = e0 + e1 + e2 + e3;
      const float p1 = e1 / sm;
      const float p2 = e2 / sm;
      int b1 = (int)(p1 * 20.0f); b1 = min(max(b1, 0), 19);
      int b2 = (int)(p2 * 20.0f); b2 = min(max(b2, 0), 19);
      atomicAdd(&h[b1], 1u);
      if (yi == 1) atomicAdd(&h[20 + b1], 1u);
      atomicAdd(&h[40 + b2], 1u);
      if (yi == 2) atomicAdd(&h[60 + b2], 1u);
    }
  }

  // Wave-level exact reduction: 5 WMMAs per stat, no LDS / shuffles.
  // Uniform control flow here: EXEC is all ones as WMMA requires.
  float st[20];
#pragma unroll
  for (int cc = 0; cc < 4; ++cc) {
    st[cc]      = sp[cc];
    st[4 + cc]  = ipr[cc];
    st[8 + cc]  = ct[cc];
    st[12 + cc] = ss[cc];
    st[16 + cc] = sps[cc];
  }
#pragma unroll
  for (int k = 0; k < 20; ++k) {
    const float r = wave_allreduce(st[k]);   // total in every lane
    if (lane == 0) wp[wid * 20 + k] = r;
  }
  __syncthreads();

  if (tid < 20) {
    float a = 0.0f;
#pragma unroll
    for (int w = 0; w < 8; ++w) a += wp[w * 20 + tid];
    blockPartials[(size_t)blockIdx.x * 20 + tid] = a;  // deterministic store
  }

  // Flush per-wave histograms with integer atomics (exact, order-free).
  for (int i = tid; i < 80; i += 256) {
    unsigned int a = 0u;
#pragma unroll
    for (int w = 0; w < 8; ++w) a += hist[w * 80 + i];
    atomicAdd(&histG[i], a);
  }
}

__global__ void finalize_kernel(const float* __restrict__ blockPartials,
                                const unsigned int* __restrict__ histG,
                                const float* __restrict__ calib,
                                const int* __restrict__ weightp,
                                float* __restrict__ out, int bpb) {
  __shared__ float tot[40];  // [batch*20 + stat]
  const int t = threadIdx.x;
  if (t < 40) {
    const int b = t / 20, s = t % 20;
    float a = 0.0f;
    for (int j = 0; j < bpb; ++j)
      a += blockPartials[(size_t)(b * bpb + j) * 20 + s];
    tot[t] = a;
  }
  __syncthreads();
  if (t == 0) {
    const float Vf = (float)V;
    const float means[4] = {0.03f, 0.02f, 0.01f, 0.01f};
    float L1 = 0.0f, L2 = 0.0f;
    for (int c = 0; c < 4; ++c) {
      const float m  = means[c];
      const float w1 = 1.0f / (m * m);
      const float w0 = 1.0f / ((1.0f - m) * (1.0f - m));
      const float sp0 = tot[c],        sp1 = tot[20 + c];
      const float ip0 = tot[4 + c],    ip1 = tot[20 + 4 + c];
      const float ct0 = tot[8 + c],    ct1 = tot[20 + 8 + c];
      const float ss0 = tot[12 + c],   ss1 = tot[20 + 12 + c];
      const float ps0 = tot[16 + c],   ps1 = tot[20 + 16 + c];
      // --- one-hot target dice ---
      {
        const float union1 = (sp0 + sp1) + (ct0 + ct1);
        const float union0 = 4.0f * Vf - union1;
        const float uni    = w0 * union0 + w1 * union1;
        float i0a = Vf - sp0 - ct0 + ip0;
        float i0b = Vf - sp1 - ct1 + ip1;
        float d0 = (2.0f * (w0 * i0a + w1 * ip0) + 1.0f) / (uni + 1.0f);
        float d1 = (2.0f * (w0 * i0b + w1 * ip1) + 1.0f) / (uni + 1.0f);
        if (isnan(d0)) d0 = 1.0f;
        if (isnan(d1)) d1 = 1.0f;
        L1 += 1.0f - 0.5f * (d0 + d1);
      }
      // --- sigmoid(logits) target dice ---
      {
        const float union1 = (sp0 + sp1) + (ss0 + ss1);
        const float union0 = 4.0f * Vf - union1;
        const float uni    = w0 * union0 + w1 * union1;
        float i0a = Vf - sp0 - ss0 + ps0;
        float i0b = Vf - sp1 - ss1 + ps1;
        float d0 = (2.0f * (w0 * i0a + w1 * ps0) + 1.0f) / (uni + 1.0f);
        float d1 = (2.0f * (w0 * i0b + w1 * ps1) + 1.0f) / (uni + 1.0f);
        if (isnan(d0)) d0 = 1.0f;
        if (isnan(d1)) d1 = 1.0f;
        L2 += 1.0f - 0.5f * (d0 + d1);
      }
    }
    L1 *= 0.2f;
    L2 *= 0.2f;
    const float w = (float)(*weightp);
    const float flip = w * L1 + (1.0f - w) * L2;

    float ece = 0.0f;
    for (int c = 1; c <= 2; ++c) {
      const unsigned int* hh = histG + (c - 1) * 40;
      float acc = 0.0f;
      for (int k = 0; k < 20; ++k) {
        const float totl = (float)hh[k];
        const float tru  = (float)hh[20 + k];
        const float ratio = sigmoidf_(tru) / sigmoidf_(totl); // empty bin -> 1
        const float cal   = sigmoidf_(calib[k * 4 + c]);
        const float d     = cal - ratio;
        acc += d * d;
      }
      ece += acc * (1.0f / 20.0f);
    }
    out[0] = flip + ece;
  }
}

extern "C" void kernel_launch(void* const* d_in, const int* in_sizes, int n_in,
                              void* d_out, int out_size, void* d_ws, size_t ws_size,
                              hipStream_t stream) {
  const float* seg    = (const float*)d_in[0];
  const float* calib  = (const float*)d_in[1];
  const float* logits = (const float*)d_in[2];
  const int*   y      = (const int*)d_in[3];
  const int*   weight = (const int*)d_in[4];
  float* out = (float*)d_out;

  // Workspace: [0..320) histogram (80 u32), [512..) block partials.
  unsigned int* histG = (unsigned int*)d_ws;
  float* blockPartials = (float*)((char*)d_ws + 512);

  // Deterministic tier selection from the (constant) workspace size.
  int bpb;  // blocks per batch
  const size_t base = 512;
  if (ws_size >= base + (size_t)NBATCH * 512 * 20 * sizeof(float)) {
    bpb = 512;   // 1024 blocks total, 4 float4-groups per thread
  } else if (ws_size >= base + (size_t)NBATCH * 256 * 20 * sizeof(float)) {
    bpb = 256;
  } else {
    bpb = 64;
  }
  const int iters = (V / 4) / (bpb * 256);  // float4 groups per thread

  zero_hist<<<1, 128, 0, stream>>>(histG);
  stats_kernel<<<NBATCH * bpb, 256, 0, stream>>>(seg, logits, y, blockPartials,
                                                 histG, bpb, iters);
  finalize_kernel<<<1, 64, 0, stream>>>(blockPartials, histG, calib, weight,
                                        out, bpb);
}